// GCN_ancestor_sequential_38981123179104
// MI455X (gfx1250) — compile-verified
//
#include <hip/hip_runtime.h>

typedef __attribute__((ext_vector_type(2))) float v2f;
typedef __attribute__((ext_vector_type(8))) float v8f;

// ---------------- degree / normalization ----------------

__global__ void deg_init_kernel(float* deg, int n) {
  int i = blockIdx.x * blockDim.x + threadIdx.x;
  if (i < n) deg[i] = 1.0f;  // self-loop contributes 1 to every node's degree
}

__global__ void deg_count_kernel(const int* __restrict__ dst, float* deg, int e) {
  int i = blockIdx.x * blockDim.x + threadIdx.x;
  if (i < e) atomicAdd(&deg[dst[i]], 1.0f);
}

__global__ void deg_rsqrt_kernel(float* deg, int n) {
  int i = blockIdx.x * blockDim.x + threadIdx.x;
  if (i < n) deg[i] = rsqrtf(deg[i]);
}

// ---------------- edge scatter-add: 64 threads per edge ----------------
// agg[d,:] += xw[s,:] * dinv[s]*dinv[d]; coalesced 256B gather per edge,
// float atomics resolve in L2 (agg is 25.6MB << 192MB L2).

__global__ void scatter_kernel(const float* __restrict__ xw, const int* __restrict__ src,
                               const int* __restrict__ dst, const float* __restrict__ dinv,
                               float* agg, int e) {
  int t = blockIdx.x * blockDim.x + threadIdx.x;
  int ed = t >> 6;
  int f  = t & 63;
  if (ed < e) {
    int s = src[ed], d = dst[ed];
    float norm = dinv[s] * dinv[d];
    atomicAdd(&agg[(size_t)d * 64 + f], xw[(size_t)s * 64 + f] * norm);
  }
}

// x = relu(agg + b)
__global__ void bias_relu_kernel(const float* __restrict__ agg, const float* __restrict__ b,
                                 float* x, int total) {
  int i = blockIdx.x * blockDim.x + threadIdx.x;
  if (i < total) {
    float v = agg[i] + b[i & 63];
    x[i] = v > 0.f ? v : 0.f;
  }
}

// ---------------- fused WMMA fp32 GEMM ----------------
// One wave computes a 16 x NC slab with V_WMMA_F32_16X16X4_F32 (K step 4).
// Weights staged in LDS *pre-interleaved* as float2 {W[k][n], W[k+1][n]} so each
// B fragment is one aligned ds_load_b64 straight into the WMMA source pair
// (no v_dual_mov repacking). Epilogue:
//   FINAL=false: out = A@W  (raw, for edge gather),  agg = out * dinv[row]^2 (self-loop seed)
//   FINAL=true : out = relu(A@W + bias)              (classifier head)
// CONCAT=true : A-matrix is [x1 | x2] (JumpingKnowledge cat), K=128.

template <int K, int NC, bool CONCAT, bool FINAL>
__global__ __launch_bounds__(256) void gemm_wmma_kernel(
    const float* __restrict__ A, const float* __restrict__ A2,
    const float* __restrict__ W, const float* __restrict__ bias,
    const float* __restrict__ dinv, float* __restrict__ out,
    float* __restrict__ agg, int nrows) {
  constexpr int NT = NC / 16;
  constexpr int KH = K / 2;
  __shared__ v2f sW[KH * NC];  // sW[kh*NC + n] = {W[2kh][n], W[2kh+1][n]}
  for (int i = threadIdx.x; i < KH * NC; i += 256) {
    int kh = i / NC, n = i - kh * NC;
    v2f w;
    w.x = W[(2 * kh) * NC + n];
    w.y = W[(2 * kh + 1) * NC + n];
    sW[i] = w;
  }
  __syncthreads();

  const int wave = threadIdx.x >> 5;
  const int lane = threadIdx.x & 31;
  const int rowBase = (blockIdx.x * 8 + wave) * 16;

  // A-fragment (16x4 f32 layout): lanes 0-15 -> K=k0,k0+1 ; lanes 16-31 -> K=k0+2,k0+3
  int mrow = rowBase + (lane & 15);
  if (mrow >= nrows) mrow = nrows - 1;        // clamp OOB loads; stores are predicated
  const int kHalf = lane >> 4;                // 0 or 1
  const int kOff  = kHalf * 2;
  const int nCol  = lane & 15;

  const v8f vzero = {0.f, 0.f, 0.f, 0.f, 0.f, 0.f, 0.f, 0.f};
  v8f acc[NT];
#pragma unroll
  for (int t = 0; t < NT; ++t) acc[t] = vzero;

  for (int k0 = 0; k0 < K; k0 += 4) {
    const int kk = k0 + kOff;                 // even, so {kk, kk+1} stay in the same 64-col half
    v2f a;
    if (CONCAT) {
      const float* sp = (kk < 64) ? (A + (size_t)mrow * 64 + kk)
                                  : (A2 + (size_t)mrow * 64 + (kk - 64));
      a = *(const v2f*)sp;                    // 8B-aligned: single global_load_b64
    } else {
      a = *(const v2f*)(A + (size_t)mrow * K + kk);
    }
    const int kPair = (k0 >> 1) + kHalf;      // float2-pair row in sW
#pragma unroll
    for (int t = 0; t < NT; ++t) {
      v2f b = sW[kPair * NC + t * 16 + nCol]; // single ds_load_b64, already packed
      acc[t] = __builtin_amdgcn_wmma_f32_16x16x4_f32(
          false, a, false, b, (short)0, acc[t], false, false);
    }
  }

  // C/D layout: VGPR j -> M=j (lanes 0-15), M=j+8 (lanes 16-31); N = lane&15 within tile
  const int rhalf = kHalf * 8;
#pragma unroll
  for (int t = 0; t < NT; ++t) {
    const int n = t * 16 + nCol;
#pragma unroll
    for (int j = 0; j < 8; ++j) {
      const int r = rowBase + rhalf + j;
      if (r < nrows) {
        float v = acc[t][j];
        if (FINAL) {
          v += bias[n];
          out[(size_t)r * NC + n] = v > 0.f ? v : 0.f;
        } else {
          out[(size_t)r * NC + n] = v;
          const float di = dinv[r];
          agg[(size_t)r * NC + n] = v * di * di;   // self-loop term seeds the aggregation
        }
      }
    }
  }
}

// ---------------- launch ----------------

extern "C" void kernel_launch(void* const* d_in, const int* in_sizes, int n_in,
                              void* d_out, int out_size, void* d_ws, size_t ws_size,
                              hipStream_t stream) {
  const float* x  = (const float*)d_in[0];
  const int*   ei = (const int*)d_in[1];
  const float* W1 = (const float*)d_in[2];
  const float* b1 = (const float*)d_in[3];
  const float* W2 = (const float*)d_in[4];
  const float* b2 = (const float*)d_in[5];
  const float* W3 = (const float*)d_in[6];
  const float* b3 = (const float*)d_in[7];
  float* out = (float*)d_out;

  const int N = in_sizes[0] / 128;
  const int E = in_sizes[1] / 2;
  const int* src = ei;        // edge_index row 0
  const int* dst = ei + E;    // edge_index row 1

  // workspace: dinv[N] | bufA[N*64] | bufB[N*64] | x1[N*64] | x2[N*64]  (~103 MB)
  float* dinv = (float*)d_ws;
  float* bufA = dinv + N;
  float* bufB = bufA + (size_t)N * 64;
  float* x1   = bufB + (size_t)N * 64;
  float* x2   = x1   + (size_t)N * 64;

  const int tpb = 256;
  const int gN  = (N + tpb - 1) / tpb;
  const int gE  = (E + tpb - 1) / tpb;
  const int gF  = (N * 64 + tpb - 1) / tpb;
  const int gS  = (int)(((long long)E * 64 + tpb - 1) / tpb);
  const int gG  = (N + 127) / 128;   // 8 waves x 16 rows per block

  // D^{-1/2}
  deg_init_kernel <<<gN, tpb, 0, stream>>>(dinv, N);
  deg_count_kernel<<<gE, tpb, 0, stream>>>(dst, dinv, E);
  deg_rsqrt_kernel<<<gN, tpb, 0, stream>>>(dinv, N);

  // layer 1: xw1=bufA, agg seed in bufB
  gemm_wmma_kernel<128, 64, false, false><<<gG, tpb, 0, stream>>>(x, nullptr, W1, nullptr, dinv, bufA, bufB, N);
  scatter_kernel  <<<gS, tpb, 0, stream>>>(bufA, src, dst, dinv, bufB, E);
  bias_relu_kernel<<<gF, tpb, 0, stream>>>(bufB, b1, x1, N * 64);

  // layer 2: xw2=bufB (reuses agg1), agg seed in bufA (reuses xw1)
  gemm_wmma_kernel<64, 64, false, false><<<gG, tpb, 0, stream>>>(x1, nullptr, W2, nullptr, dinv, bufB, bufA, N);
  scatter_kernel  <<<gS, tpb, 0, stream>>>(bufB, src, dst, dinv, bufA, E);
  bias_relu_kernel<<<gF, tpb, 0, stream>>>(bufA, b2, x2, N * 64);

  // head: out = relu([x1|x2] @ W3 + b3)
  gemm_wmma_kernel<128, 16, true, true><<<gG, tpb, 0, stream>>>(x1, x2, W3, b3, nullptr, out, nullptr, N);
}